// CacheUpdateFp8_32315333935798
// MI455X (gfx1250) — compile-verified
//
#include <hip/hip_runtime.h>

// ---------------------------------------------------------------------------
// CacheUpdateFp8 (decode branch) for MI455X / gfx1250.
//
// out = bf16(dequant_e4m3(quant_e4m3(prev_f32)))          [bulk stream, 64M elems]
// out[:, :, idx-1, :] = bf16(dequant_e4m3(quant_e4m3(cur_bf16)))  [16K elems]
//
// Memory-bound: 256MB in + 128MB out ~= 384MB => ~16.5us at 23.3 TB/s.
// Strategy: b128 NT loads -> hw fp8 pack/unpack (v_cvt_pk_fp8_f32 /
// v_cvt_pk_f32_fp8, RNE, OCP e4m3 on CDNA5) -> bf16 RNE pack -> b128 NT store.
// ---------------------------------------------------------------------------

typedef unsigned int   u32;
typedef unsigned short u16;
typedef float vf4 __attribute__((ext_vector_type(4)));
typedef float vf2 __attribute__((ext_vector_type(2)));
typedef u32   vu4 __attribute__((ext_vector_type(4)));

#if defined(__AMDGCN__) && __has_builtin(__builtin_amdgcn_cvt_pk_fp8_f32) && \
    __has_builtin(__builtin_amdgcn_cvt_pk_f32_fp8)
#define HW_FP8 1
#else
#define HW_FP8 0
#endif

// RNE f32 -> bf16 (exact for any e4m3-representable value), pack two into u32.
__device__ __forceinline__ u32 bf16pair(float a, float b) {
  u32 ua = __float_as_uint(a), ub = __float_as_uint(b);
  ua += 0x7FFFu + ((ua >> 16) & 1u);
  ub += 0x7FFFu + ((ub >> 16) & 1u);
  return (ua >> 16) | (ub & 0xFFFF0000u);
}

// Software quant-dequant through e4m3 (fallback only; host pass / no builtin).
__device__ __forceinline__ float sw_qdq_e4m3(float x) {
  float ax = fabsf(x);
  if (ax != ax) return x;            // NaN passthrough
  if (ax > 448.f) ax = 448.f;        // e4m3fn saturation
  int e;
  (void)frexpf(ax, &e);              // ax = m * 2^e, m in [0.5, 1)
  int ue = e - 1 - 3;                // exponent of the e4m3 ulp (normal range)
  if (ue < -9) ue = -9;              // denormal floor (min denorm = 2^-9)
  float unit = exp2f((float)ue);
  float q = rintf(ax / unit) * unit; // RNE quantize
  return copysignf(q, x);
}

// 8 fp32 -> e4m3 -> 8 bf16 packed in a vu4.
__device__ __forceinline__ void qdq8(vf4 x, vf4 y, vu4 &o) {
#if HW_FP8
  u32 p0 = 0, p1 = 0;
  p0 = (u32)__builtin_amdgcn_cvt_pk_fp8_f32(x.x, x.y, (int)p0, false);
  p0 = (u32)__builtin_amdgcn_cvt_pk_fp8_f32(x.z, x.w, (int)p0, true);
  p1 = (u32)__builtin_amdgcn_cvt_pk_fp8_f32(y.x, y.y, (int)p1, false);
  p1 = (u32)__builtin_amdgcn_cvt_pk_fp8_f32(y.z, y.w, (int)p1, true);
  vf2 d0 = __builtin_amdgcn_cvt_pk_f32_fp8((int)p0, false);
  vf2 d1 = __builtin_amdgcn_cvt_pk_f32_fp8((int)p0, true);
  vf2 d2 = __builtin_amdgcn_cvt_pk_f32_fp8((int)p1, false);
  vf2 d3 = __builtin_amdgcn_cvt_pk_f32_fp8((int)p1, true);
  o.x = bf16pair(d0.x, d0.y);
  o.y = bf16pair(d1.x, d1.y);
  o.z = bf16pair(d2.x, d2.y);
  o.w = bf16pair(d3.x, d3.y);
#else
  float f[8] = {x.x, x.y, x.z, x.w, y.x, y.y, y.z, y.w};
  float g[8];
  #pragma unroll
  for (int i = 0; i < 8; ++i) g[i] = sw_qdq_e4m3(f[i]);
  o.x = bf16pair(g[0], g[1]);
  o.y = bf16pair(g[2], g[3]);
  o.z = bf16pair(g[4], g[5]);
  o.w = bf16pair(g[6], g[7]);
#endif
}

// Bulk stream: each thread handles 8 contiguous fp32 -> 8 bf16.
// NT cache policy on both sides: 384MB one-touch stream > 192MB L2.
__global__ __launch_bounds__(256) void cache_qdq_stream(
    const vf4* __restrict__ in, vu4* __restrict__ out, unsigned ngrp) {
  unsigned gid = blockIdx.x * 256u + threadIdx.x;
  if (gid >= ngrp) return;
  vf4 x = __builtin_nontemporal_load(in + 2ull * gid);
  vf4 y = __builtin_nontemporal_load(in + 2ull * gid + 1);
  vu4 o;
  qdq8(x, y, o);
  __builtin_nontemporal_store(o, out + gid);
}

// Slot fixup: overwrite seq position (idx-1) with qdq(cur). One bf16 pair per
// thread; 8192 threads total. Runs after the bulk pass via stream ordering.
__global__ __launch_bounds__(256) void cache_slot_update(
    const u32* __restrict__ cur_pairs,   // cur viewed as bf16 pairs
    const int* __restrict__ idx_ptr,
    u32* __restrict__ out_pairs,         // output viewed as bf16 pairs
    unsigned npairs, unsigned pairs_per_row, unsigned seq_len) {
  unsigned t = blockIdx.x * 256u + threadIdx.x;
  if (t >= npairs) return;
  int slot = idx_ptr[0] - 1;
  unsigned row = t / pairs_per_row;      // flattened (b*H + h)
  unsigned col = t % pairs_per_row;      // pair index within D
  u32 pair = cur_pairs[t];
  float a = __uint_as_float(pair << 16);          // low bf16
  float b = __uint_as_float(pair & 0xFFFF0000u);  // high bf16
#if HW_FP8
  u32 p = (u32)__builtin_amdgcn_cvt_pk_fp8_f32(a, b, 0, false);
  vf2 d = __builtin_amdgcn_cvt_pk_f32_fp8((int)p, false);
  u32 o = bf16pair(d.x, d.y);
#else
  u32 o = bf16pair(sw_qdq_e4m3(a), sw_qdq_e4m3(b));
#endif
  size_t off = (size_t)row * seq_len * pairs_per_row +
               (size_t)slot * pairs_per_row + col;
  out_pairs[off] = o;
}

extern "C" void kernel_launch(void* const* d_in, const int* in_sizes, int n_in,
                              void* d_out, int out_size, void* d_ws, size_t ws_size,
                              hipStream_t stream) {
  const float* prev = (const float*)d_in[0];   // [B,H,S,D] fp32
  const u32*   cur  = (const u32*)d_in[1];     // [B,H,1,D] bf16 as pairs
  // d_in[2] = dim (always 2, unused)
  const int*   idx  = (const int*)d_in[3];     // scalar idx on device

  unsigned n    = (unsigned)in_sizes[0];       // 67,108,864
  unsigned ncur = (unsigned)in_sizes[1];       // 16,384 = B*H*D

  // Bulk quantize-dequantize stream: 8 elems/thread.
  unsigned ngrp = n / 8u;
  cache_qdq_stream<<<dim3((ngrp + 255u) / 256u), dim3(256), 0, stream>>>(
      (const vf4*)prev, (vu4*)d_out, ngrp);

  // Slot overwrite with qdq(cur).
  unsigned npairs = ncur / 2u;                 // 8192 bf16 pairs
  unsigned D = 128u;
  unsigned pairs_per_row = D / 2u;             // 64
  unsigned seq_len = n / ncur;                 // 4096
  cache_slot_update<<<dim3((npairs + 255u) / 256u), dim3(256), 0, stream>>>(
      cur, idx, (u32*)d_out, npairs, pairs_per_row, seq_len);
}